// SmoothAP_MC_81475529605815
// MI455X (gfx1250) — compile-verified
//
#include <hip/hip_runtime.h>
#include <hip/hip_bf16.h>

#define BATCH   64
#define NUM_ID  8
#define FEAT    256
#define GALLERY 2048
#define GROUP   8          // BATCH / NUM_ID
#define ANNEAL_INV 100.0f  // 1 / 0.01

typedef __attribute__((ext_vector_type(2))) float v2f;
typedef __attribute__((ext_vector_type(8))) float v8f;

// temperature sigmoid with exponent clamp, matching reference _tsigmoid
__device__ __forceinline__ float tsig(float x) {
    float e = fminf(fmaxf(-x * ANNEAL_INV, -50.0f), 50.0f);
    return 1.0f / (1.0f + __expf(e));
}

// ---------------------------------------------------------------------------
// Kernel 1: permute preds rows into P_perm, compute ||p_i||^2 and ||g_j||^2.
// blocks [0,64): pred rows (with camera-interleave permutation)
// blocks [64, 64+2048): gallery row norms
// ---------------------------------------------------------------------------
__global__ void __launch_bounds__(64)
prep_kernel(const float* __restrict__ preds, const float* __restrict__ gallery,
            float* __restrict__ Pp, float* __restrict__ pnorm,
            float* __restrict__ gnorm) {
    __shared__ float red[64];
    const int b = blockIdx.x, t = threadIdx.x;
    float acc = 0.0f;
    if (b < BATCH) {
        // dest row b = n*8 + c*4 + k  <-  src row c*32 + n*4 + k
        const int n = b >> 3, r = b & 7, c = r >> 2, kk = r & 3;
        const int src = c * 32 + n * 4 + kk;
        for (int idx = t; idx < FEAT; idx += 64) {
            const float v = preds[src * FEAT + idx];
            Pp[b * FEAT + idx] = v;
            acc += v * v;
        }
    } else {
        const int g = b - BATCH;
        for (int idx = t; idx < FEAT; idx += 64) {
            const float v = gallery[g * FEAT + idx];
            acc += v * v;
        }
    }
    red[t] = acc;
    __syncthreads();
    for (int s = 32; s > 0; s >>= 1) {
        if (t < s) red[t] += red[t + s];
        __syncthreads();
    }
    if (t == 0) {
        if (b < BATCH) pnorm[b] = red[0];
        else           gnorm[b - BATCH] = red[0];
    }
}

// ---------------------------------------------------------------------------
// Kernel 2: sim_all = -sqrt(max(||p||^2 + ||g||^2 - 2 p.g, 1e-12))
// One wave32 per 16x16 output tile; K-loop of V_WMMA_F32_16X16X4_F32.
// A lane layout: A[l%16][2*(l/16)+{0,1}]  -> contiguous pair of P row
// B lane layout: B[2*(l/16)+{0,1}][l%16]  -> contiguous pair of G row
// D layout: vgpr v, lane l -> C[v + 8*(l/16)][l%16]
// ---------------------------------------------------------------------------
__global__ void __launch_bounds__(32)
gemm_sim_kernel(const float* __restrict__ Pp, const float* __restrict__ G,
                const float* __restrict__ pnorm, const float* __restrict__ gnorm,
                float* __restrict__ sim) {
    const int lane = threadIdx.x;
    const int half = lane >> 4, ln = lane & 15;
    const int n0 = blockIdx.x * 16;   // gallery tile
    const int m0 = blockIdx.y * 16;   // pred tile

    const float* __restrict__ arow = Pp + (m0 + ln) * FEAT + 2 * half;
    const float* __restrict__ brow = G  + (n0 + ln) * FEAT + 2 * half;

    v8f acc = {0.f, 0.f, 0.f, 0.f, 0.f, 0.f, 0.f, 0.f};
#pragma unroll 4
    for (int k = 0; k < FEAT; k += 4) {
        const v2f a = *(const v2f*)(arow + k);
        const v2f b = *(const v2f*)(brow + k);
        acc = __builtin_amdgcn_wmma_f32_16x16x4_f32(
            /*neg_a=*/false, a, /*neg_b=*/false, b,
            /*c_mod=*/(short)0, acc, /*reuse_a=*/false, /*reuse_b=*/false);
    }

    const int col = n0 + ln;
    const float gn = gnorm[col];
#pragma unroll
    for (int v = 0; v < 8; ++v) {
        const int row = m0 + v + 8 * half;
        const float d2 = pnorm[row] + gn - 2.0f * acc[v];
        sim[row * GALLERY + col] = -sqrtf(fmaxf(d2, 1e-12f));
    }
}

// ---------------------------------------------------------------------------
// Kernel 3: blocks[row][j] = sum_k tsig(sim[row][k] - sim[row][n*8+j]),
// j = 0..7, n = row/8.  Only the diagonal 8x8 blocks of sim_all_rk are ever
// consumed by the reference, so only these 512 rank sums are computed.
// One block per row; the 8KB row lives in LDS.
// ---------------------------------------------------------------------------
__global__ void __launch_bounds__(256)
rank_kernel(const float* __restrict__ sim, float* __restrict__ blocks) {
    __shared__ float srow[GALLERY];
    __shared__ float sred[256];
    const int row = blockIdx.x, t = threadIdx.x;
    const float* __restrict__ r = sim + row * GALLERY;
    for (int k = t; k < GALLERY; k += 256) srow[k] = r[k];
    __syncthreads();

    const int n = row >> 3;
    float sj[GROUP];
#pragma unroll
    for (int j = 0; j < GROUP; ++j) sj[j] = srow[n * GROUP + j];

    float acc[GROUP];
#pragma unroll
    for (int j = 0; j < GROUP; ++j) acc[j] = 0.0f;

    for (int k = t; k < GALLERY; k += 256) {
        const float sk = srow[k];
#pragma unroll
        for (int j = 0; j < GROUP; ++j) acc[j] += tsig(sk - sj[j]);
    }

    for (int j = 0; j < GROUP; ++j) {
        __syncthreads();
        sred[t] = acc[j];
        __syncthreads();
        for (int s = 128; s > 0; s >>= 1) {
            if (t < s) sred[t] += sred[t + s];
            __syncthreads();
        }
        if (t == 0) blocks[row * GROUP + j] = sred[0];
    }
}

// ---------------------------------------------------------------------------
// Kernel 4: per-identity positive-set ranking, divided by blocks, reduced to
// one partial per identity (no float atomics -> deterministic).
// ---------------------------------------------------------------------------
__global__ void __launch_bounds__(64)
pos_kernel(const float* __restrict__ Pp, const float* __restrict__ gpos,
           const float* __restrict__ blocks, float* __restrict__ partial) {
    __shared__ float sp[64];
    __shared__ float red[64];
    const int n = blockIdx.x, t = threadIdx.x;
    const int i = t >> 3, j = t & 7;

    const float* __restrict__ x = Pp   + (n * GROUP + i) * FEAT;
    const float* __restrict__ y = gpos + (n * GROUP + j) * FEAT;
    float d2 = 0.0f;
    for (int k = 0; k < FEAT; ++k) {
        const float d = x[k] - y[k];
        d2 += d * d;
    }
    sp[t] = -sqrtf(fmaxf(d2, 1e-12f));
    __syncthreads();

    const float sij = sp[i * 8 + j];
    float rk = 0.0f;
#pragma unroll
    for (int k = 0; k < 8; ++k) rk += tsig(sp[i * 8 + k] - sij);

    red[t] = rk / blocks[(n * GROUP + i) * GROUP + j];
    __syncthreads();
    for (int s = 32; s > 0; s >>= 1) {
        if (t < s) red[t] += red[t + s];
        __syncthreads();
    }
    if (t == 0) partial[n] = red[0];
}

// ---------------------------------------------------------------------------
// Kernel 5: ordered final sum -> 1 - ap
// ---------------------------------------------------------------------------
__global__ void __launch_bounds__(32)
final_kernel(const float* __restrict__ partial, float* __restrict__ out) {
    if (threadIdx.x == 0 && blockIdx.x == 0) {
        float s = 0.0f;
        for (int n = 0; n < NUM_ID; ++n) s += partial[n];
        out[0] = 1.0f - s / (float)(GROUP * BATCH);
    }
}

extern "C" void kernel_launch(void* const* d_in, const int* in_sizes, int n_in,
                              void* d_out, int out_size, void* d_ws, size_t ws_size,
                              hipStream_t stream) {
    (void)in_sizes; (void)n_in; (void)out_size; (void)ws_size;
    const float* preds   = (const float*)d_in[0];
    const float* gallery = (const float*)d_in[1];
    const float* gpos    = (const float*)d_in[2];

    float* ws      = (float*)d_ws;
    float* Pp      = ws;                       // 64*256      = 16384 floats
    float* pnorm   = Pp + BATCH * FEAT;        // 64
    float* gnorm   = pnorm + BATCH;            // 2048
    float* sim     = gnorm + GALLERY;          // 64*2048     = 131072
    float* blocks  = sim + BATCH * GALLERY;    // 64*8        = 512
    float* partial = blocks + BATCH * GROUP;   // 8
    // total ~ 600 KB of workspace, fully rewritten each call

    prep_kernel<<<BATCH + GALLERY, 64, 0, stream>>>(preds, gallery, Pp, pnorm, gnorm);

    dim3 ggrid(GALLERY / 16, BATCH / 16);
    gemm_sim_kernel<<<ggrid, 32, 0, stream>>>(Pp, gallery, pnorm, gnorm, sim);

    rank_kernel<<<BATCH, 256, 0, stream>>>(sim, blocks);

    pos_kernel<<<NUM_ID, 64, 0, stream>>>(Pp, gpos, blocks, partial);

    final_kernel<<<1, 32, 0, stream>>>(partial, (float*)d_out);
}